// LiteratiQuantKVCache_61838939127939
// MI455X (gfx1250) — compile-verified
//
#include <hip/hip_runtime.h>

typedef float f32x4 __attribute__((ext_vector_type(4)));

// CDNA5 async global->LDS copy (VGLOBAL encoding, GV mode: 64-bit VGPR addr).
// VDST = VGPR holding LDS byte address, VADDR = VGPR pair holding global addr.
// Tracked with ASYNCcnt (CDNA5 ISA).
#define ASYNC_LOAD_B128(lds_u32, gaddr64)                                      \
  asm volatile("global_load_async_to_lds_b128 %0, %1, off"                     \
               :: "v"(lds_u32), "v"(gaddr64) : "memory")

#define WAIT_ASYNCCNT(n)                                                       \
  asm volatile("s_wait_asynccnt " #n ::: "memory")

__global__ __launch_bounds__(256)
void quant1bit_roundtrip(const float* __restrict__ x,
                         float* __restrict__ out,
                         long long nRows)
{
    // 8 waves/block * 2 buffers * 512 B (one 128-float row) = 8 KB
    __shared__ char smem[8 * 1024];

    const int wid  = threadIdx.x >> 5;   // wave id in block (wave32!)
    const int lane = threadIdx.x & 31;

    // This lane's 16-byte slot in this wave's buffer 0.
    char* ldsPtr = &smem[wid * 1024 + lane * 16];
    // Low 32 bits of the generic LDS address == LDS byte offset (ISA §10.2).
    const unsigned ldsAddr = (unsigned)(size_t)ldsPtr;

    const long long stride = (long long)gridDim.x * 8;
    long long r = (long long)blockIdx.x * 8 + wid;

    int par = 0;
    if (r < nRows) {
        const float* gp = x + r * 128 + lane * 4;
        ASYNC_LOAD_B128(ldsAddr, (unsigned long long)(size_t)gp);
    }
    while (r < nRows) {
        const long long rn = r + stride;
        if (rn < nRows) {
            // Prefetch next row into the other buffer, then wait for current.
            const float* gp = x + rn * 128 + lane * 4;
            ASYNC_LOAD_B128(ldsAddr + (unsigned)((par ^ 1) * 512),
                            (unsigned long long)(size_t)gp);
            WAIT_ASYNCCNT(1);   // async loads retire in order: row r is in LDS
        } else {
            WAIT_ASYNCCNT(0);
        }

        // Pull this lane's 4 elements of the row from LDS (ds_load_b128).
        const f32x4 v = *(const f32x4*)(ldsPtr + par * 512);

        float s2 = v.x * v.x + v.y * v.y + v.z * v.z + v.w * v.w;
        float sa = __builtin_fabsf(v.x) + __builtin_fabsf(v.y) +
                   __builtin_fabsf(v.z) + __builtin_fabsf(v.w);

        // wave32 butterfly reduction: 5 steps, result broadcast to all lanes.
        #pragma unroll
        for (int m = 16; m >= 1; m >>= 1) {
            s2 += __shfl_xor(s2, m, 32);
            sa += __shfl_xor(sa, m, 32);
        }

        float norm = __builtin_sqrtf(s2);
        norm = __builtin_fmaxf(norm, 1e-8f);              // clip(norm, EPS)
        // scale = mean(|x|/norm) rounded through fp16 (RTNE), back to f32
        const float scale = (float)(_Float16)(sa / (norm * 128.0f));
        const float a = scale * norm;                     // per-element magnitude

        f32x4 o;
        o.x = (v.x > 0.0f) ? a : -a;                      // bit = (x_unit > 0)
        o.y = (v.y > 0.0f) ? a : -a;
        o.z = (v.z > 0.0f) ? a : -a;
        o.w = (v.w > 0.0f) ? a : -a;

        // Streaming store, non-temporal: no reuse, keep L2 clean.
        __builtin_nontemporal_store(o, (f32x4*)(out + r * 128 + lane * 4));

        par ^= 1;
        r = rn;
    }
}

extern "C" void kernel_launch(void* const* d_in, const int* in_sizes, int n_in,
                              void* d_out, int out_size, void* d_ws, size_t ws_size,
                              hipStream_t stream) {
    (void)n_in; (void)out_size; (void)d_ws; (void)ws_size;
    const float* x = (const float*)d_in[0];
    float* out = (float*)d_out;

    const long long nElem = (long long)in_sizes[0];
    const long long nRows = nElem / 128;                  // 524,288 for the ref shapes

    // 8 waves/block, target ~16 rows per wave -> 4096 blocks for the ref shapes.
    long long blocks = (nRows + 8 * 16 - 1) / (8 * 16);
    if (blocks < 1) blocks = 1;
    if (blocks > 65535) blocks = 65535;

    quant1bit_roundtrip<<<(int)blocks, 256, 0, stream>>>(x, out, nRows);
}